// TransBlock_13924283973827
// MI455X (gfx1250) — compile-verified
//
#include <hip/hip_runtime.h>
#include <hip/hip_bf16.h>
#include <math.h>

// ---------------------------------------------------------------------------
// TransBlock for MI455X (gfx1250, wave32).  All contractions use the native
// fp32 matrix instruction V_WMMA_F32_16X16X4_F32 (exact fp32 vs the fp32
// reference).  One wave = one 16x16 output tile.
// Round 3: one-operand masking (zero-side absorbs garbage-side), scale folded
// into Q, float4 softmax, zero per-iteration masking VALU in all WMMA loops.
// ---------------------------------------------------------------------------

typedef float v2f __attribute__((ext_vector_type(2)));
typedef float v4f __attribute__((ext_vector_type(4)));
typedef float v8f __attribute__((ext_vector_type(8)));

#define DIM   288
#define RED   144
#define HEADS 8
#define HD    18
#define NCHNK 4
#define NPOS  4096            // 64*64 per batch
#define BATCH 4
#define MROWS (BATCH * NPOS)  // 16384
#define CHUNK 1024
#define QKVLD 432             // 3*RED row stride of fused qkv buffer
#define ATT_SCALE 0.2357022603955158f   // 18^-0.5

static __device__ __forceinline__ v8f wmma_f32(v2f a, v2f b, v8f c) {
    // D = A(16x4, f32) x B(4x16, f32) + C(16x16, f32)
    return __builtin_amdgcn_wmma_f32_16x16x4_f32(false, a, false, b,
                                                 (short)0, c, false, false);
}

static __device__ __forceinline__ v2f ld2(const float* p) {
    return *(const v2f*)p;   // even float offset -> 8B aligned -> b64 load
}

// ---------------------------------------------------------------------------
// 1) 3x3 patch extraction (torch.nn.Unfold ordering): t[b, n, c*9+ki*3+kj]
// ---------------------------------------------------------------------------
__global__ __launch_bounds__(256)
void patch_kernel(const float* __restrict__ x, float* __restrict__ t) {
    int gid = blockIdx.x * 256 + threadIdx.x;
    const int total = MROWS * DIM;
    if (gid >= total) return;
    int d = gid % DIM;
    int n = (gid / DIM) % NPOS;
    int b = gid / (DIM * NPOS);
    int c  = d / 9;
    int k  = d % 9;
    int ki = k / 3 - 1;
    int kj = k % 3 - 1;
    int i  = n >> 6;
    int j  = n & 63;
    int ii = i + ki;
    int jj = j + kj;
    float v = 0.0f;
    if (ii >= 0 && ii < 64 && jj >= 0 && jj < 64)
        v = x[(((size_t)b * 32 + c) * 64 + ii) * 64 + jj];
    t[gid] = v;
}

// ---------------------------------------------------------------------------
// 2) LayerNorm over last dim (288), one wave per row
// ---------------------------------------------------------------------------
__global__ __launch_bounds__(32)
void ln_kernel(const float* __restrict__ in, const float* __restrict__ g,
               const float* __restrict__ be, float* __restrict__ outp) {
    int row = blockIdx.x;
    int l   = threadIdx.x;
    const float* p = in + (size_t)row * DIM;
    float s = 0.0f, s2 = 0.0f;
    #pragma unroll
    for (int i = 0; i < 9; ++i) {
        float v = p[l + 32 * i];
        s  += v;
        s2 += v * v;
    }
    for (int off = 16; off > 0; off >>= 1) {
        s  += __shfl_xor(s,  off, 32);
        s2 += __shfl_xor(s2, off, 32);
    }
    float mu  = s  * (1.0f / DIM);
    float var = s2 * (1.0f / DIM) - mu * mu;
    float inv = rsqrtf(var + 1e-5f);
    float* o = outp + (size_t)row * DIM;
    #pragma unroll
    for (int i = 0; i < 9; ++i) {
        int idx = l + 32 * i;
        o[idx] = (p[idx] - mu) * inv * g[idx] + be[idx];
    }
}

// ---------------------------------------------------------------------------
// 3) Generic fp32 WMMA GEMM: C = act(A[M,K] @ W[K,N] + bias) (+ residual)
//    One wave per 16x16 tile; K multiple of 4 (288/144/72 all are).
//    N-edge: clamped column only -- garbage columns are never stored, so no
//    masking VALU in the K loop at all.
// ---------------------------------------------------------------------------
template <bool RELU>
__global__ __launch_bounds__(32)
void wgemm_kernel(const float* __restrict__ A, const float* __restrict__ W,
                  const float* __restrict__ bias, const float* __restrict__ res,
                  float* __restrict__ C, int M, int K, int N) {
    int m0   = blockIdx.x * 16;
    int n0   = blockIdx.y * 16;
    int l    = threadIdx.x;
    int half = l >> 4;      // 0: K rows k0,k0+1   1: K rows k0+2,k0+3
    int lr   = l & 15;      // A: M index, B: N index

    v8f acc = {};
    const float* arow = A + (size_t)(m0 + lr) * K;
    int  nb  = n0 + lr;
    bool nok = (nb < N);
    int  nbc = nok ? nb : (N - 1);    // clamped, always a valid address

    __builtin_prefetch(arow, 0, 0);                       // global_prefetch_b8
    __builtin_prefetch(W + (size_t)(2 * half) * N + nbc, 0, 0);

    #pragma unroll 2
    for (int k0 = 0; k0 < K; k0 += 4) {
        int ka = k0 + 2 * half;
        v2f a = ld2(arow + ka);                           // global_load_b64
        v2f b;
        b.x = W[(size_t)ka * N + nbc];
        b.y = W[(size_t)(ka + 1) * N + nbc];
        acc = wmma_f32(a, b, acc);
    }

    if (!nok) return;
    float bv = bias ? bias[nb] : 0.0f;
    #pragma unroll
    for (int r = 0; r < 8; ++r) {
        int mr = m0 + r + 8 * half;
        float v = acc[r] + bv;
        if (RELU) v = v > 0.0f ? v : 0.0f;
        if (res)  v += res[(size_t)mr * N + nb];
        C[(size_t)mr * N + nb] = v;
    }
}

// ---------------------------------------------------------------------------
// 4) Chunked attention.  One wave per (b, head, chunk, 16-row q-tile).
//    Scores strip 16x1024 in LDS (64 KB of the 320 KB WGP LDS), float4
//    softmax in-wave, AV via WMMA.
//    HD=18 padded to 20; only the Q (A-side) fragment is masked -- zeroed
//    A k-slices annihilate the K-matrix's garbage k=18,19 rows, and garbage
//    B columns only feed never-stored outputs.  Scale folded into Q.
// ---------------------------------------------------------------------------
__global__ __launch_bounds__(32)
void attn_kernel(const float* __restrict__ qkv, float* __restrict__ out) {
    __shared__ float sc[16 * CHUNK];   // 64 KB

    int id = blockIdx.x;
    int qt = id & 63;        id >>= 6;   // q-tile within chunk
    int cc = id & 3;         id >>= 2;   // chunk
    int hh = id & 7;         id >>= 3;   // head
    int b  = id;                          // batch

    int l    = threadIdx.x;
    int half = l >> 4;
    int lr   = l & 15;

    const float* qbase = qkv + (size_t)(b * NPOS + cc * CHUNK) * QKVLD + hh * HD;
    const float* kbase = qbase + RED;
    const float* vbase = qbase + 2 * RED;

    // pair d = 4*kk + 2*half: fully valid for kk<4; kk==4 valid only half==0
    float s4 = half ? 0.0f : ATT_SCALE;   // mask*scale for last Q pair
    int   d4 = half ? 0 : 16;             // clamped offset for last pair

    // ---- Q tile into A-registers (pre-scaled, pad-masked on A side only) ----
    v2f qa[5];
    const float* qrow = qbase + (size_t)(qt * 16 + lr) * QKVLD;
    #pragma unroll
    for (int kk = 0; kk < 4; ++kk) {
        v2f q2 = ld2(qrow + 4 * kk + 2 * half);
        qa[kk].x = q2.x * ATT_SCALE;
        qa[kk].y = q2.y * ATT_SCALE;
    }
    {
        v2f q2 = ld2(qrow + d4);
        qa[4].x = q2.x * s4;
        qa[4].y = q2.y * s4;
    }

    // ---- S = (Q*scale) K^T  -> LDS strip (raw b64 K loads, no masking) ----
    for (int kt = 0; kt < CHUNK / 16; ++kt) {
        v8f acc = {};
        const float* krow = kbase + (size_t)(kt * 16 + lr) * QKVLD;
        #pragma unroll
        for (int kk = 0; kk < 4; ++kk)
            acc = wmma_f32(qa[kk], ld2(krow + 4 * kk + 2 * half), acc);
        acc = wmma_f32(qa[4], ld2(krow + d4), acc);
        #pragma unroll
        for (int r = 0; r < 8; ++r)
            sc[(r + 8 * half) * CHUNK + kt * 16 + lr] = acc[r];
    }
    __syncthreads();

    // ---- softmax: 2 lanes per row, 512 columns each, float4 passes ----
    {
        int row = l >> 1;
        int seg = l & 1;
        v4f* p = (v4f*)(sc + row * CHUNK + seg * 512);
        v4f mx4 = p[0];
        for (int i = 1; i < 128; ++i) {
            v4f v = p[i];
            mx4.x = fmaxf(mx4.x, v.x); mx4.y = fmaxf(mx4.y, v.y);
            mx4.z = fmaxf(mx4.z, v.z); mx4.w = fmaxf(mx4.w, v.w);
        }
        float mx = fmaxf(fmaxf(mx4.x, mx4.y), fmaxf(mx4.z, mx4.w));
        mx = fmaxf(mx, __shfl_xor(mx, 1, 32));
        v4f sum4 = {};
        for (int i = 0; i < 128; ++i) {
            v4f v = p[i];
            v.x = __expf(v.x - mx); v.y = __expf(v.y - mx);
            v.z = __expf(v.z - mx); v.w = __expf(v.w - mx);
            p[i] = v;
            sum4 += v;
        }
        float sum = (sum4.x + sum4.y) + (sum4.z + sum4.w);
        sum += __shfl_xor(sum, 1, 32);
        float inv = 1.0f / sum;
        for (int i = 0; i < 128; ++i) {
            v4f v = p[i];
            v.x *= inv; v.y *= inv; v.z *= inv; v.w *= inv;
            p[i] = v;
        }
    }
    __syncthreads();

    // ---- O = P V : K=1024 WMMA loop, two N-tiles (cols 0-15, 16-17) ----
    // tile1 lanes lr>=2 read clamped col 16 -> garbage outputs never stored
    v8f o0 = {}, o1 = {};
    bool hi   = (lr < HD - 16);
    int  col2 = 16 + (hi ? lr : 0);
    const float* srow = sc + lr * CHUNK;
    #pragma unroll 4
    for (int k0 = 0; k0 < CHUNK; k0 += 4) {
        int ka = k0 + 2 * half;
        v2f a = ld2(srow + ka);                     // ds_load_b64
        const float* v0 = vbase + (size_t)ka * QKVLD;
        const float* v1 = v0 + QKVLD;
        v2f b0, b1;
        b0.x = v0[lr];
        b0.y = v1[lr];
        b1.x = v0[col2];
        b1.y = v1[col2];
        o0 = wmma_f32(a, b0, o0);
        o1 = wmma_f32(a, b1, o1);
    }

    // ---- write attn_out[b, n, head*18 + d]  (RED=144 row stride) ----
    float* ob = out + (size_t)(b * NPOS + cc * CHUNK) * RED + hh * HD;
    #pragma unroll
    for (int r = 0; r < 8; ++r) {
        int qr = qt * 16 + r + 8 * half;
        ob[(size_t)qr * RED + lr] = o0[r];
        if (hi) ob[(size_t)qr * RED + 16 + lr] = o1[r];
    }
}

// ---------------------------------------------------------------------------
// Driver
// ---------------------------------------------------------------------------
extern "C" void kernel_launch(void* const* d_in, const int* in_sizes, int n_in,
                              void* d_out, int out_size, void* d_ws, size_t ws_size,
                              hipStream_t stream) {
    const float* x        = (const float*)d_in[0];
    const float* n1g      = (const float*)d_in[1];
    const float* n1b      = (const float*)d_in[2];
    const float* reduce_w = (const float*)d_in[3];
    const float* qkv_w    = (const float*)d_in[4];
    const float* proj_w   = (const float*)d_in[5];
    const float* proj_b   = (const float*)d_in[6];
    const float* n2g      = (const float*)d_in[7];
    const float* n2b      = (const float*)d_in[8];
    const float* fc1_w    = (const float*)d_in[9];
    const float* fc1_b    = (const float*)d_in[10];
    const float* fc2_w    = (const float*)d_in[11];
    const float* fc2_b    = (const float*)d_in[12];
    float*       out      = (float*)d_out;

    // workspace layout (floats); total ~103.5 MB
    float* ws   = (float*)d_ws;
    float* t    = ws;                                 // [16384, 288]
    float* ln   = t   + (size_t)MROWS * DIM;          // [16384, 288] (reused)
    float* h    = ln  + (size_t)MROWS * DIM;          // [16384, 144]
    float* qkv  = h   + (size_t)MROWS * RED;          // [16384, 432]
    float* ao   = qkv + (size_t)MROWS * 3 * RED;      // [16384, 144]
    float* t2   = ao  + (size_t)MROWS * RED;          // [16384, 288]
    float* mlp  = t2  + (size_t)MROWS * DIM;          // [16384, 72]

    // 1) patch extraction -> t
    patch_kernel<<<(MROWS * DIM + 255) / 256, 256, 0, stream>>>(x, t);
    // 2) ln1
    ln_kernel<<<MROWS, 32, 0, stream>>>(t, n1g, n1b, ln);
    // 3) reduce: [16384,288]x[288,144]
    wgemm_kernel<false><<<dim3(MROWS / 16, RED / 16), 32, 0, stream>>>(
        ln, reduce_w, nullptr, nullptr, h, MROWS, DIM, RED);
    // 4) qkv: [16384,144]x[144,432]
    wgemm_kernel<false><<<dim3(MROWS / 16, (3 * RED) / 16), 32, 0, stream>>>(
        h, qkv_w, nullptr, nullptr, qkv, MROWS, RED, 3 * RED);
    // 5) attention: 4*8*4 chunks * 64 q-tiles = 8192 waves
    attn_kernel<<<BATCH * HEADS * NCHNK * (CHUNK / 16), 32, 0, stream>>>(qkv, ao);
    // 6) proj + residual(t): [16384,144]x[144,288]
    wgemm_kernel<false><<<dim3(MROWS / 16, DIM / 16), 32, 0, stream>>>(
        ao, proj_w, proj_b, t, t2, MROWS, RED, DIM);
    // 7) ln2
    ln_kernel<<<MROWS, 32, 0, stream>>>(t2, n2g, n2b, ln);
    // 8) fc1 + relu: [16384,288]x[288,72]  (N=72 -> 5 n-tiles, clamped edge)
    wgemm_kernel<true><<<dim3(MROWS / 16, (DIM / 4 + 15) / 16), 32, 0, stream>>>(
        ln, fc1_w, fc1_b, nullptr, mlp, MROWS, DIM, DIM / 4);
    // 9) fc2 + residual(t2) -> out: [16384,72]x[72,288]
    wgemm_kernel<false><<<dim3(MROWS / 16, DIM / 16), 32, 0, stream>>>(
        mlp, fc2_w, fc2_b, t2, out, MROWS, DIM / 4, DIM);

    (void)in_sizes; (void)n_in; (void)out_size; (void)ws_size;
}